// ProposalLayer_53747220742218
// MI455X (gfx1250) — compile-verified
//
#include <hip/hip_runtime.h>
#include <hip/hip_bf16.h>
#include <stdint.h>

// ---------------- problem constants ----------------
#define BATCH 8
#define NA    262144          // anchors per batch (power of two)
#define KSEL  6000            // PRE_NMS_LIMIT
#define PROP  1000            // PROPOSAL_COUNT
#define NMS_THR 0.7f
#define NBINS 4096
#define CAP   2048            // capacity for threshold-bin candidates
#define EPB   8192            // elements per block for hist/compact (256 thr * 32)
#define BPB   (NA / EPB)      // blocks per batch = 32
#define NEGF  (-3.0e38f)
#define PBATCH_F 30000        // floats per batch in staging: 6000*4 boxes + 6000 scores
#define NMS_LDS_BYTES 120320  // 96000 boxes + 24000 scores + 320 reduce scratch

typedef unsigned int u32;
typedef __attribute__((ext_vector_type(4))) u32 v4u;
typedef __attribute__((ext_vector_type(4))) int  v4i;
typedef __attribute__((ext_vector_type(8))) int  v8i;

#if defined(__has_builtin)
#if __has_builtin(__builtin_amdgcn_tensor_load_to_lds)
#define HAVE_TDM 1
#endif
#endif
// Only fail if the *device* pass lacks the TDM builtin (host pass never has it).
#if defined(__HIP_DEVICE_COMPILE__) && !defined(HAVE_TDM)
#error "gfx1250 TDM builtin __builtin_amdgcn_tensor_load_to_lds not available in device pass"
#endif

// ---------------- kernel 1: zero hist + counters ----------------
__global__ void k_zero(u32* p, int n) {
    int i = blockIdx.x * blockDim.x + threadIdx.x;
    if (i < n) p[i] = 0u;
}

__device__ __forceinline__ int score_bin(float s) {
    s = fminf(fmaxf(s, 0.0f), 1.0f);
    int b = (int)(s * (float)NBINS);
    return b > (NBINS - 1) ? (NBINS - 1) : b;
}

// ---------------- kernel 2: per-batch score histogram (LDS privatized) ----------------
__global__ void k_hist(const float* __restrict__ probs, u32* __restrict__ hist) {
    __shared__ u32 lh[NBINS];
    for (int i = threadIdx.x; i < NBINS; i += blockDim.x) lh[i] = 0u;
    __syncthreads();
    const int b = blockIdx.x / BPB;
    const int chunk = blockIdx.x % BPB;
    const size_t base = (size_t)b * NA + (size_t)chunk * EPB;
    const float2* p2 = (const float2*)probs;      // [.x,.y] = [prob0, prob1]
    for (int t = threadIdx.x; t < EPB; t += blockDim.x) {
        // prefetch next stripe of this block's tile (global_prefetch_b8; speculative, OOB-safe)
        __builtin_prefetch(&p2[base + t + 1024], 0, 1);
        float s = p2[base + t].y;
        atomicAdd(&lh[score_bin(s)], 1u);
    }
    __syncthreads();
    u32* gh = hist + (size_t)b * NBINS;
    for (int i = threadIdx.x; i < NBINS; i += blockDim.x)
        if (lh[i]) atomicAdd(&gh[i], lh[i]);
}

// ---------------- kernel 3: find threshold bin per batch ----------------
// cut[b] = {T, countAbove, need, 0}
__global__ void k_cut(const u32* __restrict__ hist, u32* __restrict__ cut) {
    if (threadIdx.x != 0) return;
    const int b = blockIdx.x;
    const u32* h = hist + (size_t)b * NBINS;
    u32 acc = 0, T = 0, above = 0, need = KSEL;
    for (int bin = NBINS - 1; bin >= 0; --bin) {
        u32 c = h[bin];
        if (acc + c >= (u32)KSEL) { T = (u32)bin; above = acc; need = (u32)KSEL - acc; break; }
        acc += c;
    }
    cut[b * 4 + 0] = T; cut[b * 4 + 1] = above; cut[b * 4 + 2] = need; cut[b * 4 + 3] = 0u;
}

// ---------------- kernel 4: compaction ----------------
__global__ void k_compact(const float* __restrict__ probs, const u32* __restrict__ cut,
                          u32* __restrict__ cnts, u32* __restrict__ sel,
                          float* __restrict__ candS, u32* __restrict__ candI) {
    const int b = blockIdx.x / BPB;
    const int chunk = blockIdx.x % BPB;
    const int T = (int)cut[b * 4 + 0];
    const size_t base = (size_t)b * NA + (size_t)chunk * EPB;
    const float2* p2 = (const float2*)probs;
    for (int t = threadIdx.x; t < EPB; t += blockDim.x) {
        __builtin_prefetch(&p2[base + t + 1024], 0, 1);
        const u32 a = (u32)(chunk * EPB + t);
        float s = p2[base + t].y;
        int bin = score_bin(s);
        if (bin > T) {
            u32 pos = atomicAdd(&cnts[b * 4 + 0], 1u);
            if (pos < (u32)KSEL) sel[(size_t)b * KSEL + pos] = a;
        } else if (bin == T) {
            u32 pos = atomicAdd(&cnts[b * 4 + 1], 1u);
            if (pos < (u32)CAP) { candS[(size_t)b * CAP + pos] = s; candI[(size_t)b * CAP + pos] = a; }
        }
    }
}

// ---------------- kernel 5: bitonic select top-`need` within threshold bin ----------------
__global__ void k_sel(const u32* __restrict__ cut, const u32* __restrict__ cnts,
                      const float* __restrict__ candS, const u32* __restrict__ candI,
                      u32* __restrict__ sel) {
    __shared__ float key[CAP];
    __shared__ u32   kid[CAP];
    const int b = blockIdx.x;
    const u32 m = min(cnts[b * 4 + 1], (u32)CAP);
    const u32 above = cut[b * 4 + 1];
    const u32 need = cut[b * 4 + 2];
    for (int i = threadIdx.x; i < CAP; i += blockDim.x) {
        bool v = (u32)i < m;
        key[i] = v ? -candS[(size_t)b * CAP + i] : 3.0e38f;  // ascending sort of -score
        kid[i] = v ? candI[(size_t)b * CAP + i] : 0u;
    }
    __syncthreads();
    for (int k2 = 2; k2 <= CAP; k2 <<= 1) {
        for (int j = k2 >> 1; j > 0; j >>= 1) {
            for (int i = threadIdx.x; i < CAP; i += blockDim.x) {
                int ixj = i ^ j;
                if (ixj > i) {
                    bool up = ((i & k2) == 0);
                    float a0 = key[i], a1 = key[ixj];
                    if (up ? (a0 > a1) : (a0 < a1)) {
                        key[i] = a1; key[ixj] = a0;
                        u32 t0 = kid[i]; kid[i] = kid[ixj]; kid[ixj] = t0;
                    }
                }
            }
            __syncthreads();
        }
    }
    const u32 take = min(need, m);
    for (u32 t = threadIdx.x; t < need; t += blockDim.x)
        sel[(size_t)b * KSEL + above + t] = (t < take) ? kid[t] : 0u;
}

// ---------------- kernel 6: gather + apply deltas + clip, stage [boxes|scores] ----------------
__global__ void k_mat(const float* __restrict__ probs, const float* __restrict__ bbox,
                      const float* __restrict__ anch, const u32* __restrict__ sel,
                      float* __restrict__ bx) {
    int t = blockIdx.x * blockDim.x + threadIdx.x;
    if (t >= BATCH * KSEL) return;
    const int b = t / KSEL, k = t % KSEL;
    const u32 i = sel[t] & (u32)(NA - 1);
    const size_t gi = (size_t)b * NA + i;
    float4 an = ((const float4*)anch)[gi];
    float4 dl = ((const float4*)bbox)[gi];
    float dy = dl.x * 0.1f, dx = dl.y * 0.1f, dh = dl.z * 0.2f, dw = dl.w * 0.2f;
    float h = an.z - an.x, w = an.w - an.y;
    float cy = an.x + 0.5f * h + dy * h;
    float cx = an.y + 0.5f * w + dx * w;
    h *= expf(dh);
    w *= expf(dw);
    float4 o;
    o.x = fminf(fmaxf(cy - 0.5f * h, 0.f), 1.f);
    o.y = fminf(fmaxf(cx - 0.5f * w, 0.f), 1.f);
    o.z = fminf(fmaxf(cy + 0.5f * h, 0.f), 1.f);
    o.w = fminf(fmaxf(cx + 0.5f * w, 0.f), 1.f);
    float* dst = bx + (size_t)b * PBATCH_F;
    ((float4*)dst)[k] = o;
    dst[4 * KSEL + k] = ((const float2*)probs)[gi].y;   // score
}

// ---------------- kernel 7: serial greedy NMS, one workgroup per batch ----------------
// 120 KB working set staged in LDS via TDM (tensor_load_to_lds + s_wait_tensorcnt).
__global__ void __launch_bounds__(1024) k_nms(const float* __restrict__ bx, float* __restrict__ out) {
    extern __shared__ __align__(16) char smem[];
    float4* sBox = (float4*)smem;                 // 6000 * 16 = 96000 B
    float*  sAct = (float*)(smem + 96000);        // 6000 * 4  = 24000 B
    float*  rV   = (float*)(smem + 120000);       // 40 floats
    int*    rI   = (int*)(smem + 120160);         // 40 ints
    const int tid = threadIdx.x;
    const int b = blockIdx.x;
    const float* src = bx + (size_t)b * PBATCH_F;

#ifdef HAVE_TDM
    if (tid < 32) {  // one wave issues the DMA (TDM ignores EXEC; guard per-wave)
        unsigned long long ga = (unsigned long long)(uintptr_t)src;
        u32 lds_off = (u32)__builtin_amdgcn_groupstaticsize();  // dynamic LDS base
        v4u g0;
        g0.x = 1u;                                   // count=1, user mode
        g0.y = lds_off;                              // lds_addr
        g0.z = (u32)(ga & 0xFFFFFFFFull);            // global_addr[31:0]
        g0.w = (u32)((ga >> 32) & 0x1FFFFFFull) | (2u << 30);  // addr[56:32] | type=2
        v8i g1;
        g1[0] = (int)(2u << 16);                     // data_size = 4 B
        g1[1] = (int)((u32)(PBATCH_F & 0xFFFF) << 16);          // tensor_dim0 lo
        g1[2] = (int)(((u32)PBATCH_F >> 16) | (1u << 16));      // tensor_dim0 hi | tensor_dim1=1
        g1[3] = (int)((u32)(PBATCH_F & 0xFFFF) << 16);          // tile_dim0 = 30000
        g1[4] = 1;                                   // tile_dim1 = 1
        g1[5] = PBATCH_F;                            // tensor_dim0_stride lo
        g1[6] = 0;
        g1[7] = 0;
        v4i g2 = {0, 0, 0, 0};
        v4i g3 = {0, 0, 0, 0};
#if defined(__clang_major__) && (__clang_major__ >= 23)
        v8i gx = {0, 0, 0, 0, 0, 0, 0, 0};
        __builtin_amdgcn_tensor_load_to_lds(g0, g1, g2, g3, gx, 0);
#else
        __builtin_amdgcn_tensor_load_to_lds(g0, g1, g2, g3, 0);
#endif
        __builtin_amdgcn_s_wait_tensorcnt(0);
    }
    __syncthreads();
#else
    // host-pass / non-TDM parse path only
    for (int i = tid; i < PBATCH_F; i += 1024) ((float*)smem)[i] = src[i];
    __syncthreads();
#endif

    const int wid = tid >> 5, lane = tid & 31;
    for (int it = 0; it < PROP; ++it) {
        // block-wide argmax over sAct (smallest index wins ties)
        float bv = NEGF; int bi = 0;
        for (int j = tid; j < KSEL; j += 1024) {
            float s = sAct[j];
            if (s > bv) { bv = s; bi = j; }
        }
        for (int off = 16; off > 0; off >>= 1) {
            float ov = __shfl_down(bv, off, 32);
            int   oi = __shfl_down(bi, off, 32);
            if (ov > bv || (ov == bv && oi < bi)) { bv = ov; bi = oi; }
        }
        if (lane == 0) { rV[wid] = bv; rI[wid] = bi; }
        __syncthreads();
        if (wid == 0) {
            bv = rV[lane]; bi = rI[lane];
            for (int off = 16; off > 0; off >>= 1) {
                float ov = __shfl_down(bv, off, 32);
                int   oi = __shfl_down(bi, off, 32);
                if (ov > bv || (ov == bv && oi < bi)) { bv = ov; bi = oi; }
            }
            if (lane == 0) { rV[32] = bv; rI[32] = bi; }
        }
        __syncthreads();
        const float pv = rV[32];
        const int pick = rI[32];
        if (pv > -1.0e30f) {
            float4 pb = sBox[pick];
            if (tid == 0) ((float4*)out)[b * PROP + it] = pb;
            float pa = (pb.z - pb.x) * (pb.w - pb.y);
            for (int j = tid; j < KSEL; j += 1024) {
                if (sAct[j] > -1.0e30f) {
                    float4 q = sBox[j];
                    float yy1 = fmaxf(pb.x, q.x);
                    float xx1 = fmaxf(pb.y, q.y);
                    float yy2 = fminf(pb.z, q.z);
                    float xx2 = fminf(pb.w, q.w);
                    float inter = fmaxf(yy2 - yy1, 0.f) * fmaxf(xx2 - xx1, 0.f);
                    float uni = pa + (q.z - q.x) * (q.w - q.y) - inter;
                    if (inter > NMS_THR * fmaxf(uni, 1e-12f)) sAct[j] = NEGF;
                }
            }
            if (tid == 0) sAct[pick] = NEGF;
        } else if (tid == 0) {
            float4 z = {0.f, 0.f, 0.f, 0.f};
            ((float4*)out)[b * PROP + it] = z;
        }
        __syncthreads();
    }
}

// ---------------- launch ----------------
extern "C" void kernel_launch(void* const* d_in, const int* in_sizes, int n_in,
                              void* d_out, int out_size, void* d_ws, size_t ws_size,
                              hipStream_t stream) {
    (void)in_sizes; (void)n_in; (void)out_size; (void)ws_size;
    const float* probs = (const float*)d_in[0];   // (B, A, 2)
    const float* bbox  = (const float*)d_in[1];   // (B, A, 4)
    const float* anch  = (const float*)d_in[2];   // (B, A, 4)
    float* out = (float*)d_out;                   // (B, PROP, 4)

    // workspace carve (byte offsets, all 16B-aligned)
    char* ws = (char*)d_ws;
    u32*   hist  = (u32*)(ws + 0);                                   // B*NBINS      -> 131072 B
    u32*   cnts  = (u32*)(ws + 131072);                              // B*4          -> 128 B
    u32*   cut   = (u32*)(ws + 131200);                              // B*4          -> 128 B
    u32*   sel   = (u32*)(ws + 131328);                              // B*KSEL       -> 192000 B
    float* candS = (float*)(ws + 323328);                            // B*CAP        -> 65536 B
    u32*   candI = (u32*)(ws + 388864);                              // B*CAP        -> 65536 B
    float* bx    = (float*)(ws + 454400);                            // B*30000 f32  -> 960000 B

    const int nzero = BATCH * NBINS + BATCH * 4;  // hist + counters
    k_zero<<<(nzero + 255) / 256, 256, 0, stream>>>((u32*)ws, nzero);
    k_hist<<<BATCH * BPB, 256, 0, stream>>>(probs, hist);
    k_cut<<<BATCH, 32, 0, stream>>>(hist, cut);
    k_compact<<<BATCH * BPB, 256, 0, stream>>>(probs, cut, cnts, sel, candS, candI);
    k_sel<<<BATCH, 256, 0, stream>>>(cut, cnts, candS, candI, sel);
    k_mat<<<(BATCH * KSEL + 255) / 256, 256, 0, stream>>>(probs, bbox, anch, sel, bx);

    (void)hipFuncSetAttribute((const void*)k_nms,
                              hipFuncAttributeMaxDynamicSharedMemorySize,
                              NMS_LDS_BYTES);
    k_nms<<<BATCH, 1024, NMS_LDS_BYTES, stream>>>(bx, out);
}